// VectorKuramoto_27848567947911
// MI455X (gfx1250) — compile-verified
//
#include <hip/hip_runtime.h>

typedef __attribute__((ext_vector_type(2))) float v2f;
typedef __attribute__((ext_vector_type(8))) float v8f;

#define BB 4
#define NN 2048
#define DD 8
#define TILE_M 16
#define WAVES 8                    // 256 threads = 8 wave32
#define K_PER_WAVE (NN / WAVES)    // 256
#define STEPS (K_PER_WAVE / 4)     // 64 WMMA K-steps of 4 per wave

#define COUPLING_C 1.0f
#define DT_C 1.0f
#define ATTR_C 1.0f

// ---------------------------------------------------------------------------
// Kernel 1: pack SC[b][j][0:8] = sin(theta[b][j][:]), SC[b][j][8:16] = cos(...)
// SC is the shared B-operand for all four logical matmuls. 512 KB -> L2-hot.
// ---------------------------------------------------------------------------
__global__ __launch_bounds__(256) void VK_sc_pack(const float* __restrict__ theta,
                                                  float* __restrict__ sc) {
    int idx = blockIdx.x * blockDim.x + threadIdx.x;   // over B*N*D = 65536
    if (idx >= BB * NN * DD) return;
    int d  = idx & 7;
    int bn = idx >> 3;                                 // b*N + j
    float s, c;
    __sincosf(theta[idx], &s, &c);
    sc[bn * 16 + d]     = s;
    sc[bn * 16 + d + 8] = c;
}

// ---------------------------------------------------------------------------
// Kernel 2: one workgroup per 16-row output tile (512 tiles).
// Each wave owns a disjoint K-range, accumulates Cc = Wc@SC and Cs = Ws@SC
// with V_WMMA_F32_16X16X4_F32, then partials are reduced through LDS.
// ---------------------------------------------------------------------------
__global__ __launch_bounds__(256) void VK_wmma_main(
    const float* __restrict__ theta,
    const float* __restrict__ gam,
    const float* __restrict__ aff,
    const float* __restrict__ alp,
    const float* __restrict__ sc,
    float* __restrict__ out)
{
    __shared__ float lds_part[WAVES * 2 * TILE_M * 16];   // 16 KB

    const int tile = blockIdx.x;          // 0..511
    const int b    = tile >> 7;           // tile / (N/16)
    const int i0   = (tile & 127) << 4;   // row-tile base

    const int tid  = threadIdx.x;
    const int wave = tid >> 5;
    const int lane = tid & 31;
    const int m    = lane & 15;           // A row / B col / C col index
    const int g    = lane >> 4;           // K-half selector (0/1)

    const float* __restrict__ aff_row = aff + (size_t)(b * NN + i0 + m) * NN;
    const float* __restrict__ alp_row = alp + (size_t)(b * NN + i0 + m) * NN;
    const float* __restrict__ sc_b    = sc  + (size_t)b * NN * 16;

    v8f acc_c = {};   // accumulates Wc @ [s|c]
    v8f acc_s = {};   // accumulates Ws @ [s|c]

    const int kbase = wave * K_PER_WAVE;

    for (int st = 0; st < STEPS; ++st) {
        const int j0 = kbase + st * 4;
        const int jc = j0 + 2 * g;        // this lane's two K columns

        if ((st & 7) == 0) {              // uniform branch; stream-ahead hint
            __builtin_prefetch(aff_row + jc + 32, 0, 1);
            __builtin_prefetch(alp_row + jc + 32, 0, 1);
        }

        // --- A fragments: affinity * {cos,sin}(alpha), 8B contiguous/lane ---
        v2f a2 = *(const v2f*)(aff_row + jc);
        v2f l2 = *(const v2f*)(alp_row + jc);
        float s0, c0, s1, c1;
        __sincosf(l2.x, &s0, &c0);
        __sincosf(l2.y, &s1, &c1);
        v2f wcf = { a2.x * c0, a2.y * c1 };   // v0 <- K=2g, v1 <- K=2g+1
        v2f wsf = { a2.x * s0, a2.y * s1 };

        // --- B fragment: SC rows jc, jc+1 at column m (L2 resident) ---
        v2f bf;
        bf.x = sc_b[(size_t)jc * 16 + m];
        bf.y = sc_b[(size_t)(jc + 1) * 16 + m];

        // D = A x B + C  (16x16x4, f32)
        acc_c = __builtin_amdgcn_wmma_f32_16x16x4_f32(
            false, wcf, false, bf, (short)0, acc_c, false, false);
        acc_s = __builtin_amdgcn_wmma_f32_16x16x4_f32(
            false, wsf, false, bf, (short)0, acc_s, false, false);
    }

    // --- stash per-wave partial tiles: lane holds C[r + 8*g][m] in acc[r] ---
    #pragma unroll
    for (int r = 0; r < 8; ++r) {
        const int row = r + 8 * g;
        lds_part[((wave * 2 + 0) * TILE_M + row) * 16 + m] = acc_c[r];
        lds_part[((wave * 2 + 1) * TILE_M + row) * 16 + m] = acc_s[r];
    }
    __syncthreads();

    // --- cross-wave reduction + fused epilogue: 128 outputs per tile ---
    if (tid < TILE_M * DD) {
        const int row = tid >> 3;
        const int d   = tid & 7;
        float cc_s = 0.f, cc_c = 0.f, cs_s = 0.f, cs_c = 0.f;
        #pragma unroll
        for (int w = 0; w < WAVES; ++w) {
            cc_s += lds_part[((w * 2 + 0) * TILE_M + row) * 16 + d];      // Wc@s
            cc_c += lds_part[((w * 2 + 0) * TILE_M + row) * 16 + d + 8];  // Wc@c
            cs_s += lds_part[((w * 2 + 1) * TILE_M + row) * 16 + d];      // Ws@s
            cs_c += lds_part[((w * 2 + 1) * TILE_M + row) * 16 + d + 8];  // Ws@c
        }
        const int   i  = i0 + row;
        const float si = sc_b[(size_t)i * 16 + d];
        const float ci = sc_b[(size_t)i * 16 + d + 8];
        const float coupling = (COUPLING_C / (float)NN) *
                               (ci * cc_s - si * cc_c - (ci * cs_c + si * cs_s));
        const size_t gi = (size_t)(b * NN + i) * DD + d;
        const float th = theta[gi];
        const float ga = gam[gi];
        out[gi] = th + DT_C * (ATTR_C * (ga - th) + coupling);
    }
}

// ---------------------------------------------------------------------------
extern "C" void kernel_launch(void* const* d_in, const int* in_sizes, int n_in,
                              void* d_out, int out_size, void* d_ws, size_t ws_size,
                              hipStream_t stream) {
    (void)in_sizes; (void)n_in; (void)out_size; (void)ws_size;
    const float* theta = (const float*)d_in[0];   // [B,N,D]
    const float* gam   = (const float*)d_in[1];   // [B,N,D]
    const float* aff   = (const float*)d_in[2];   // [B,N,N]
    const float* alp   = (const float*)d_in[3];   // [B,N,N]
    float*       out   = (float*)d_out;           // [B,N,D]
    float*       sc    = (float*)d_ws;            // [B,N,16] = 512 KB scratch

    const int total = BB * NN * DD;
    VK_sc_pack<<<(total + 255) / 256, 256, 0, stream>>>(theta, sc);

    const int tiles = BB * (NN / TILE_M);         // 512 workgroups
    VK_wmma_main<<<tiles, WAVES * 32, 0, stream>>>(theta, gam, aff, alp, sc, out);
}